// GATModel_49563922596646
// MI455X (gfx1250) — compile-verified
//
#include <hip/hip_runtime.h>
#include <hip/hip_bf16.h>
#include <math.h>

#define Nn 50000
#define Ee 500000
#define Fd 256
#define EDd 64
#define Hd 128
#define Od 32
#define NEG_SLOPE 0.2f
#define EPSC 1e-16f

typedef float v2f __attribute__((ext_vector_type(2)));
typedef float v8f __attribute__((ext_vector_type(8)));

// ---------------------------------------------------------------- utilities

__global__ void fill_kernel(float* __restrict__ p, float v, long long n) {
    long long i = (long long)blockIdx.x * blockDim.x + threadIdx.x;
    if (i < n) p[i] = v;
}

__device__ __forceinline__ void atomicMaxFloat(float* addr, float val) {
    // IEEE ordering trick; addr must be initialized to -INF.
    if (val >= 0.0f)
        atomicMax((int*)addr, __float_as_int(val));
    else
        atomicMin((unsigned int*)addr, (unsigned int)__float_as_int(val));
}

// ------------------------------------------------- WMMA f32 GEMM: C = A @ B
// One block = one 16-row strip of C; one wave per 16-column tile.
// M multiple of 16, K multiple of 4 (<= Fd), Ncol = 16 * (blockDim.x/32).
// A-slab (16 x K, contiguous since lda==K) is staged in LDS once and
// shared by all waves; fragments use V_WMMA_F32_16X16X4_F32 layouts
// (ISA 7.12.2): A lanes 0-15 = rows, V0/V1 = K{0,1} lower / K{2,3} upper.
__global__ void wmma_gemm_lds(const float* __restrict__ A,
                              const float* __restrict__ B,
                              float* __restrict__ C,
                              int K, int Ncol) {
    __shared__ float As[16 * Fd];            // up to 16 KB of the 320 KB WGP LDS
    const int lane = threadIdx.x & 31;
    const int wave = threadIdx.x >> 5;
    const int row0 = blockIdx.x << 4;

    // cooperative contiguous stage of the 16xK A-slab (b128 loads)
    const float4* gsrc = (const float4*)(A + (long long)row0 * K);
    float4* lds4 = (float4*)As;
    const int nvec = (16 * K) >> 2;
    for (int i = threadIdx.x; i < nvec; i += blockDim.x)
        lds4[i] = gsrc[i];
    __syncthreads();

    const int col0 = wave << 4;
    const int m  = lane & 15;                // row (A) / col (B,C) within tile
    const int kb = (lane >> 4) << 1;         // 0 for lanes 0-15, 2 for 16-31

    v8f acc = {};
    const float2* arow = (const float2*)(As + m * K);   // (k+kb) always even
    for (int k = 0; k < K; k += 4) {
        v2f a, b;
        float2 av = arow[(k + kb) >> 1];     // ds_load_b64
        a.x = av.x;
        a.y = av.y;
        b.x = B[(long long)(k + kb)     * Ncol + col0 + m];
        b.y = B[(long long)(k + kb + 1) * Ncol + col0 + m];
        acc = __builtin_amdgcn_wmma_f32_16x16x4_f32(
            /*neg_a=*/false, a, /*neg_b=*/false, b,
            /*c_mod=*/(short)0, acc, /*reuse_a=*/false, /*reuse_b=*/false);
    }
    const int rbase = row0 + ((lane >> 4) << 3);   // +8 rows for upper half-wave
    const int cc = col0 + m;
#pragma unroll
    for (int r = 0; r < 8; ++r)
        C[(long long)(rbase + r) * Ncol + cc] = acc[r];
}

// ---------------------------------------- We @ att_edge vectors (both layers)
__global__ void wvec_kernel(const float* __restrict__ We1, const float* __restrict__ ae1v,
                            const float* __restrict__ We2, const float* __restrict__ ae2v,
                            float* __restrict__ wv1, float* __restrict__ wv2) {
    int d = threadIdx.x;
    if (d < EDd) {
        float s1 = 0.f, s2 = 0.f;
        for (int j = 0; j < Hd; ++j) s1 += We1[d * Hd + j] * ae1v[j];
        for (int j = 0; j < Od; ++j) s2 += We2[d * Od + j] * ae2v[j];
        wv1[d] = s1;
        wv2[d] = s2;
    }
}

// -------- per-edge a_e for both layers + per-dst count and a_e segment sums
// one wave per edge; each lane handles 2 of the 64 edge features
__global__ void edge_ae_kernel(const int* __restrict__ dst,
                               const float* __restrict__ ea,
                               const float* __restrict__ wv1,
                               const float* __restrict__ wv2,
                               float* __restrict__ ae1, float* __restrict__ ae2,
                               float* __restrict__ sum1, float* __restrict__ sum2,
                               float* __restrict__ cnt) {
    int widx = (int)(((long long)blockIdx.x * blockDim.x + threadIdx.x) >> 5);
    int lane = threadIdx.x & 31;
    if (widx >= Ee) return;
    const float2 v  = ((const float2*)(ea + (long long)widx * EDd))[lane];
    const float2 w1 = ((const float2*)wv1)[lane];
    const float2 w2 = ((const float2*)wv2)[lane];
    float p1 = v.x * w1.x + v.y * w1.y;
    float p2 = v.x * w2.x + v.y * w2.y;
    for (int o = 16; o > 0; o >>= 1) {
        p1 += __shfl_down(p1, o, 32);
        p2 += __shfl_down(p2, o, 32);
    }
    if (lane == 0) {
        ae1[widx] = p1;
        ae2[widx] = p2;
        int d = dst[widx];
        atomicAdd(&sum1[d], p1);
        atomicAdd(&sum2[d], p2);
        atomicAdd(&cnt[d], 1.0f);
    }
}

// aeloop[n] = sum[n] / max(cnt[n], 1)   (in place on sum1/sum2)
__global__ void aeloop_kernel(float* __restrict__ sum1, float* __restrict__ sum2,
                              const float* __restrict__ cnt) {
    int n = blockIdx.x * blockDim.x + threadIdx.x;
    if (n >= Nn) return;
    float c = fmaxf(cnt[n], 1.0f);
    sum1[n] /= c;
    sum2[n] /= c;
}

// a_s[n] = h[n]·att_src ; a_d[n] = h[n]·att_dst    — one wave per node
__global__ void asd_kernel(const float* __restrict__ h,
                           const float* __restrict__ att_s,
                           const float* __restrict__ att_d,
                           float* __restrict__ a_s, float* __restrict__ a_d, int C) {
    int n = (int)(((long long)blockIdx.x * blockDim.x + threadIdx.x) >> 5);
    int lane = threadIdx.x & 31;
    if (n >= Nn) return;
    const float* row = h + (long long)n * C;
    float s = 0.f, d = 0.f;
    for (int c = lane; c < C; c += 32) {
        float v = row[c];
        s += v * att_s[c];
        d += v * att_d[c];
    }
    for (int o = 16; o > 0; o >>= 1) {
        s += __shfl_down(s, o, 32);
        d += __shfl_down(d, o, 32);
    }
    if (lane == 0) {
        a_s[n] = s;
        a_d[n] = d;
    }
}

// alpha = leaky_relu(a_s[src]+a_d[dst]+a_e); segment max via atomics
__global__ void alpha_kernel(const int* __restrict__ src, const int* __restrict__ dst,
                             const float* __restrict__ ae, const float* __restrict__ aeloop,
                             const float* __restrict__ a_s, const float* __restrict__ a_d,
                             float* __restrict__ alpha, float* __restrict__ maxb) {
    int e = blockIdx.x * blockDim.x + threadIdx.x;
    if (e >= Ee + Nn) return;
    int s, d;
    float aev;
    if (e < Ee) { s = src[e]; d = dst[e]; aev = ae[e]; }
    else        { s = d = e - Ee;        aev = aeloop[s]; }
    float v = a_s[s] + a_d[d] + aev;
    v = (v > 0.f) ? v : NEG_SLOPE * v;
    alpha[e] = v;
    atomicMaxFloat(&maxb[d], v);
}

// ex = exp(alpha - max[dst]) (in place), denom[dst] += ex
__global__ void exp_kernel(const int* __restrict__ dst,
                           float* __restrict__ alpha,
                           const float* __restrict__ maxb,
                           float* __restrict__ denom) {
    int e = blockIdx.x * blockDim.x + threadIdx.x;
    if (e >= Ee + Nn) return;
    int d = (e < Ee) ? dst[e] : (e - Ee);
    float ex = expf(alpha[e] - maxb[d]);
    alpha[e] = ex;
    atomicAdd(&denom[d], ex);
}

// acc[dst] += (ex/(denom[dst]+eps)) * h[src]   — one wave per edge
__global__ void scatter_kernel(const int* __restrict__ src, const int* __restrict__ dst,
                               const float* __restrict__ ex, const float* __restrict__ denom,
                               const float* __restrict__ h, float* __restrict__ acc, int C) {
    int widx = (int)(((long long)blockIdx.x * blockDim.x + threadIdx.x) >> 5);
    int lane = threadIdx.x & 31;
    if (widx >= Ee + Nn) return;
    int s, d;
    if (widx < Ee) { s = src[widx]; d = dst[widx]; }
    else           { s = d = widx - Ee; }
    float coef = ex[widx] / (denom[d] + EPSC);
    const float* hrow = h + (long long)s * C;
    float* arow = acc + (long long)d * C;
    for (int c = lane; c < C; c += 32)
        atomicAdd(&arow[c], coef * hrow[c]);
}

// hout = relu(acc + b)
__global__ void relu_bias_kernel(const float* __restrict__ acc, const float* __restrict__ b,
                                 float* __restrict__ hout, int C) {
    long long i = (long long)blockIdx.x * blockDim.x + threadIdx.x;
    if (i >= (long long)Nn * C) return;
    float v = acc[i] + b[(int)(i % C)];
    hout[i] = v > 0.f ? v : 0.f;
}

// out += b (final bias, no relu)
__global__ void bias_kernel(float* __restrict__ out, const float* __restrict__ b, int C) {
    long long i = (long long)blockIdx.x * blockDim.x + threadIdx.x;
    if (i >= (long long)Nn * C) return;
    out[i] += b[(int)(i % C)];
}

// ---------------------------------------------------------------------------

static inline int cdiv(long long a, long long b) { return (int)((a + b - 1) / b); }

extern "C" void kernel_launch(void* const* d_in, const int* in_sizes, int n_in,
                              void* d_out, int out_size, void* d_ws, size_t ws_size,
                              hipStream_t stream) {
    const float* x        = (const float*)d_in[0];
    const int*   eidx     = (const int*)  d_in[1];
    const float* ea       = (const float*)d_in[2];
    const float* W1       = (const float*)d_in[3];
    const float* att_src1 = (const float*)d_in[4];
    const float* att_dst1 = (const float*)d_in[5];
    const float* We1      = (const float*)d_in[6];
    const float* att_e1   = (const float*)d_in[7];
    const float* b1       = (const float*)d_in[8];
    const float* W2       = (const float*)d_in[9];
    const float* att_src2 = (const float*)d_in[10];
    const float* att_dst2 = (const float*)d_in[11];
    const float* We2      = (const float*)d_in[12];
    const float* att_e2   = (const float*)d_in[13];
    const float* b2       = (const float*)d_in[14];

    const int* src = eidx;
    const int* dst = eidx + Ee;

    // d_out layout: (out [N*O], h [N*H])
    float* outC = (float*)d_out;
    float* hout = outC + (long long)Nn * Od;

    // workspace carve-up (floats)
    float* W = (float*)d_ws;
    long long off = 0;
    float* h1    = W + off; off += (long long)Nn * Hd;
    float* acc1  = W + off; off += (long long)Nn * Hd;
    float* h2    = W + off; off += (long long)Nn * Od;
    float* a_s1  = W + off; off += Nn;
    float* a_d1  = W + off; off += Nn;
    float* a_s2  = W + off; off += Nn;
    float* a_d2  = W + off; off += Nn;
    float* cnt   = W + off; off += Nn;
    float* maxb  = W + off; off += Nn;
    float* denom = W + off; off += Nn;
    float* sum1  = W + off; off += Nn;   // becomes aeloop1
    float* sum2  = W + off; off += Nn;   // becomes aeloop2
    float* ae1   = W + off; off += Ee;
    float* ae2   = W + off; off += Ee;
    float* alpha = W + off; off += (long long)(Ee + Nn);
    float* wv1   = W + off; off += EDd;
    float* wv2   = W + off; off += EDd;

    const int BT = 256;
    const long long E2 = (long long)Ee + Nn;

    // ---- init
    fill_kernel<<<cdiv(Nn, BT), BT, 0, stream>>>(cnt, 0.f, Nn);
    fill_kernel<<<cdiv(Nn, BT), BT, 0, stream>>>(sum1, 0.f, Nn);
    fill_kernel<<<cdiv(Nn, BT), BT, 0, stream>>>(sum2, 0.f, Nn);
    fill_kernel<<<cdiv((long long)Nn * Hd, BT), BT, 0, stream>>>(acc1, 0.f, (long long)Nn * Hd);
    fill_kernel<<<cdiv((long long)Nn * Od, BT), BT, 0, stream>>>(outC, 0.f, (long long)Nn * Od);
    fill_kernel<<<cdiv(Nn, BT), BT, 0, stream>>>(maxb, -INFINITY, Nn);
    fill_kernel<<<cdiv(Nn, BT), BT, 0, stream>>>(denom, 0.f, Nn);

    // ---- edge attention precompute (reads edge_attr exactly once)
    wvec_kernel<<<1, 64, 0, stream>>>(We1, att_e1, We2, att_e2, wv1, wv2);
    edge_ae_kernel<<<cdiv((long long)Ee * 32, BT), BT, 0, stream>>>(
        dst, ea, wv1, wv2, ae1, ae2, sum1, sum2, cnt);
    aeloop_kernel<<<cdiv(Nn, BT), BT, 0, stream>>>(sum1, sum2, cnt);

    // ================= layer 1 =================
    // h1 = x @ W1 : 3125 strips x 8 column-tile waves (256 threads)
    wmma_gemm_lds<<<Nn / 16, 32 * (Hd / 16), 0, stream>>>(x, W1, h1, Fd, Hd);
    asd_kernel<<<cdiv((long long)Nn * 32, BT), BT, 0, stream>>>(h1, att_src1, att_dst1, a_s1, a_d1, Hd);
    alpha_kernel<<<cdiv(E2, BT), BT, 0, stream>>>(src, dst, ae1, sum1, a_s1, a_d1, alpha, maxb);
    exp_kernel<<<cdiv(E2, BT), BT, 0, stream>>>(dst, alpha, maxb, denom);
    scatter_kernel<<<cdiv(E2 * 32, BT), BT, 0, stream>>>(src, dst, alpha, denom, h1, acc1, Hd);
    relu_bias_kernel<<<cdiv((long long)Nn * Hd, BT), BT, 0, stream>>>(acc1, b1, hout, Hd);

    // ---- re-init segment buffers for layer 2
    fill_kernel<<<cdiv(Nn, BT), BT, 0, stream>>>(maxb, -INFINITY, Nn);
    fill_kernel<<<cdiv(Nn, BT), BT, 0, stream>>>(denom, 0.f, Nn);

    // ================= layer 2 =================
    // h2 = hout @ W2 : 3125 strips x 2 column-tile waves (64 threads)
    wmma_gemm_lds<<<Nn / 16, 32 * (Od / 16), 0, stream>>>(hout, W2, h2, Hd, Od);
    asd_kernel<<<cdiv((long long)Nn * 32, BT), BT, 0, stream>>>(h2, att_src2, att_dst2, a_s2, a_d2, Od);
    alpha_kernel<<<cdiv(E2, BT), BT, 0, stream>>>(src, dst, ae2, sum2, a_s2, a_d2, alpha, maxb);
    exp_kernel<<<cdiv(E2, BT), BT, 0, stream>>>(dst, alpha, maxb, denom);
    scatter_kernel<<<cdiv(E2 * 32, BT), BT, 0, stream>>>(src, dst, alpha, denom, h2, outC, Od);
    bias_kernel<<<cdiv((long long)Nn * Od, BT), BT, 0, stream>>>(outC, b2, Od);

    (void)in_sizes; (void)n_in; (void)out_size; (void)ws_size;
}